// MambaByte_20693152432816
// MI455X (gfx1250) — compile-verified
//
#include <hip/hip_runtime.h>
#include <stdint.h>

#define D_MODEL 1792
#define D_INNER 3584
#define D_STATE 16
#define DT_RANK 112
#define DBL_N   144
#define SEQ_L   512
#define NLAYERS 4

typedef __attribute__((ext_vector_type(2))) float v2f;
typedef __attribute__((ext_vector_type(4))) float v4f;
typedef __attribute__((ext_vector_type(8))) float v8f;

// ---------------------------------------------------------------------------
// Embedding gather: resid[l, :] = embed[ids[l], :]
// ---------------------------------------------------------------------------
__global__ __launch_bounds__(256) void embed_kernel(const int* __restrict__ ids,
                                                    const float* __restrict__ emb,
                                                    float* __restrict__ resid) {
  int l = blockIdx.x;
  int tok = ids[l];
  #pragma unroll
  for (int i = 0; i < 7; ++i) {
    int d = threadIdx.x + i * 256;
    resid[(size_t)l * D_MODEL + d] = emb[(size_t)tok * D_MODEL + d];
  }
}

// ---------------------------------------------------------------------------
// Fused (residual += add) + LayerNorm.  One block per token, 256 threads.
// ---------------------------------------------------------------------------
__global__ __launch_bounds__(256) void add_ln_kernel(float* __restrict__ resid,
                                                     const float* __restrict__ add,
                                                     const float* __restrict__ w,
                                                     const float* __restrict__ b,
                                                     float* __restrict__ out) {
  int l = blockIdx.x;
  float v[7];
  float s = 0.f, s2 = 0.f;
  #pragma unroll
  for (int i = 0; i < 7; ++i) {
    int d = threadIdx.x + i * 256;
    float x = resid[(size_t)l * D_MODEL + d];
    if (add) {
      x += add[(size_t)l * D_MODEL + d];
      resid[(size_t)l * D_MODEL + d] = x;
    }
    v[i] = x;
    s += x;
    s2 += x * x;
  }
  __shared__ float rs[8], rs2[8];
  #pragma unroll
  for (int o = 16; o > 0; o >>= 1) {
    s  += __shfl_down(s,  o);
    s2 += __shfl_down(s2, o);
  }
  int wv = threadIdx.x >> 5, ln = threadIdx.x & 31;
  if (ln == 0) { rs[wv] = s; rs2[wv] = s2; }
  __syncthreads();
  if (threadIdx.x == 0) {
    float ts = 0.f, ts2 = 0.f;
    #pragma unroll
    for (int i = 0; i < 8; ++i) { ts += rs[i]; ts2 += rs2[i]; }
    rs[0] = ts; rs2[0] = ts2;
  }
  __syncthreads();
  float mu   = rs[0] * (1.0f / D_MODEL);
  float var  = rs2[0] * (1.0f / D_MODEL) - mu * mu;
  float rstd = rsqrtf(var + 1e-5f);
  #pragma unroll
  for (int i = 0; i < 7; ++i) {
    int d = threadIdx.x + i * 256;
    out[(size_t)l * D_MODEL + d] = (v[i] - mu) * rstd * w[d] + b[d];
  }
}

// ---------------------------------------------------------------------------
// WMMA fp32 GEMM: C[M x N] = A[M x K] * W[N x K]^T   (all row-major fp32)
// Block: 256 threads (8 wave32), tile 128x128, K-block 16.
// Double-buffered LDS filled with GLOBAL_LOAD_ASYNC_TO_LDS_B128 (ASYNCcnt),
// so global->LDS traffic bypasses VGPRs and overlaps with the WMMAs.
// Each wave: 32(M) x 64(N) = 2x4 array of 16x16x4 f32 WMMA accumulators.
// ---------------------------------------------------------------------------
#define BM 128
#define BN 128
#define KB 16
#define KPAD 20   // row stride 80B: 16B-aligned B128 LDS writes, conflict-free reads

#define ASYNC_B128(lds, goff, base)                                          \
  asm volatile("global_load_async_to_lds_b128 %0, %1, %2"                    \
               :: "v"(lds), "v"(goff), "s"(base) : "memory")

__global__ __launch_bounds__(256) void gemm_wmma(const float* __restrict__ A, int lda,
                                                 const float* __restrict__ W, int ldw,
                                                 float* __restrict__ C, int ldc,
                                                 int N, int K, int nbound) {
  __shared__ float As[2][BM * KPAD];
  __shared__ float Ws[2][BN * KPAD];

  const int tid  = threadIdx.x;
  const int bm   = blockIdx.y * BM;
  const int bn   = blockIdx.x * BN;
  const int wave = tid >> 5;
  const int lane = tid & 31;
  const int hlf  = lane >> 4;
  const int r    = lane & 15;
  const int wm   = (wave & 3) * 32;
  const int wn   = (wave >> 2) * 64;

  v8f acc[2][4];
  #pragma unroll
  for (int i = 0; i < 2; ++i)
    #pragma unroll
    for (int j = 0; j < 4; ++j)
      acc[i][j] = (v8f){0.f, 0.f, 0.f, 0.f, 0.f, 0.f, 0.f, 0.f};

  const int loadRow = tid >> 2;        // 0..63
  const int loadCol = (tid & 3) << 2;  // 0,4,8,12 (floats)

  // Global byte offsets for this thread's two A rows / two W rows (k0 = 0).
  const unsigned long long baseA = (unsigned long long)(uintptr_t)A;
  const unsigned long long baseW = (unsigned long long)(uintptr_t)W;
  unsigned gA0 = ((unsigned)(bm + loadRow)      * (unsigned)lda + loadCol) * 4u;
  unsigned gA1 = ((unsigned)(bm + loadRow + 64) * (unsigned)lda + loadCol) * 4u;
  int n0 = bn + loadRow, n1 = bn + loadRow + 64;
  if (nbound) {               // clamp (not mask): uniform async-op count per wave,
    n0 = min(n0, N - 1);      // garbage only lands in never-stored columns
    n1 = min(n1, N - 1);
  }
  unsigned gW0 = ((unsigned)n0 * (unsigned)ldw + loadCol) * 4u;
  unsigned gW1 = ((unsigned)n1 * (unsigned)ldw + loadCol) * 4u;

  // LDS byte addresses (flat-pointer low 32 bits == LDS offset on AMDGPU).
  unsigned lA0[2], lA1[2], lW0[2], lW1[2];
  #pragma unroll
  for (int bfi = 0; bfi < 2; ++bfi) {
    lA0[bfi] = (unsigned)(uintptr_t)&As[bfi][loadRow * KPAD + loadCol];
    lA1[bfi] = (unsigned)(uintptr_t)&As[bfi][(loadRow + 64) * KPAD + loadCol];
    lW0[bfi] = (unsigned)(uintptr_t)&Ws[bfi][loadRow * KPAD + loadCol];
    lW1[bfi] = (unsigned)(uintptr_t)&Ws[bfi][(loadRow + 64) * KPAD + loadCol];
  }

  // Prologue: fill buffer 0.
  ASYNC_B128(lA0[0], gA0, baseA);
  ASYNC_B128(lA1[0], gA1, baseA);
  ASYNC_B128(lW0[0], gW0, baseW);
  ASYNC_B128(lW1[0], gW1, baseW);

  int cur = 0;
  for (int k0 = 0; k0 < K; k0 += KB) {
    if (k0 + KB < K) {
      const int nxt = cur ^ 1;
      const unsigned go = (unsigned)(k0 + KB) * 4u;
      ASYNC_B128(lA0[nxt], gA0 + go, baseA);
      ASYNC_B128(lA1[nxt], gA1 + go, baseA);
      ASYNC_B128(lW0[nxt], gW0 + go, baseW);
      ASYNC_B128(lW1[nxt], gW1 + go, baseW);
      // 8 outstanding; loads retire in order -> <=4 means current buffer landed.
      asm volatile("s_wait_asynccnt 0x4" ::: "memory");
    } else {
      asm volatile("s_wait_asynccnt 0x0" ::: "memory");
    }
    __syncthreads();   // everyone's current-buffer data visible

    const float* Ab = As[cur];
    const float* Wb = Ws[cur];
    #pragma unroll
    for (int kk = 0; kk < KB; kk += 4) {
      const int c = kk + 2 * hlf;
      v2f af[2], bf[4];
      #pragma unroll
      for (int i = 0; i < 2; ++i)
        af[i] = *(const v2f*)&Ab[(wm + 16 * i + r) * KPAD + c];
      #pragma unroll
      for (int j = 0; j < 4; ++j)
        bf[j] = *(const v2f*)&Wb[(wn + 16 * j + r) * KPAD + c];
      #pragma unroll
      for (int i = 0; i < 2; ++i)
        #pragma unroll
        for (int j = 0; j < 4; ++j)
          acc[i][j] = __builtin_amdgcn_wmma_f32_16x16x4_f32(
              false, af[i], false, bf[j], (short)0, acc[i][j], false, false);
    }
    __syncthreads();   // done reading 'cur'; next iteration may overwrite it
    cur ^= 1;
  }

  // Writeback.  C/D layout: VGPR v -> row (v + 8*half), lanes 0-15 -> N cols.
  #pragma unroll
  for (int i = 0; i < 2; ++i) {
    #pragma unroll
    for (int j = 0; j < 4; ++j) {
      int n = bn + wn + 16 * j + r;
      if (nbound && n >= N) continue;
      #pragma unroll
      for (int v = 0; v < 8; ++v) {
        int m = bm + wm + 16 * i + v + 8 * hlf;
        C[(size_t)m * ldc + n] = acc[i][j][v];
      }
    }
  }
}

// ---------------------------------------------------------------------------
// Depthwise causal conv (width 4) + SiLU.  xz holds [xc | z] with stride 7168.
// ---------------------------------------------------------------------------
__global__ __launch_bounds__(256) void conv_silu_kernel(const float* __restrict__ xz,
                                                        const float* __restrict__ cw,
                                                        const float* __restrict__ cb,
                                                        float* __restrict__ xc) {
  int d = blockIdx.x * 256 + threadIdx.x;   // 0..3583
  int l = blockIdx.y;                       // 0..511
  v4f wv = *(const v4f*)(cw + (size_t)d * 4);
  float acc = cb[d];
  const float w4[4] = {wv.x, wv.y, wv.z, wv.w};
  #pragma unroll
  for (int j = 0; j < 4; ++j) {
    int lj = l - 3 + j;
    if (lj >= 0) acc += w4[j] * xz[(size_t)lj * (2 * D_INNER) + d];
  }
  float sig = 1.f / (1.f + __expf(-acc));
  xc[(size_t)l * D_INNER + d] = acc * sig;
}

// ---------------------------------------------------------------------------
// Selective scan.  16 lanes per channel (one lane per state), sequential in L.
// Fused: softplus(dt+bias), dA=exp(dt*A), h update, C-reduction over states,
// D-skip, SiLU(z) gating.
// ---------------------------------------------------------------------------
__global__ __launch_bounds__(256) void scan_kernel(const float* __restrict__ dtr,
                                                   const float* __restrict__ dt_b,
                                                   const float* __restrict__ dblp,
                                                   const float* __restrict__ xc,
                                                   const float* __restrict__ xz,
                                                   const float* __restrict__ A_log,
                                                   const float* __restrict__ Dp,
                                                   float* __restrict__ y) {
  int s = threadIdx.x & 15;
  int d = blockIdx.x * 16 + (threadIdx.x >> 4);
  float Av   = -__expf(A_log[(size_t)d * D_STATE + s]);
  float bias = dt_b[d];
  float Dv   = Dp[d];
  float h = 0.f;
  for (int t = 0; t < SEQ_L; ++t) {
    float dtv = dtr[(size_t)t * D_INNER + d] + bias;
    dtv = (dtv > 20.f) ? dtv : log1pf(__expf(dtv));       // softplus
    float x  = xc[(size_t)t * D_INNER + d];
    float Bv = dblp[(size_t)t * DBL_N + DT_RANK + s];
    float Cv = dblp[(size_t)t * DBL_N + DT_RANK + D_STATE + s];
    float dA = __expf(dtv * Av);
    h = fmaf(dA, h, dtv * Bv * x);
    float p = h * Cv;
    p += __shfl_xor(p, 8, 16);
    p += __shfl_xor(p, 4, 16);
    p += __shfl_xor(p, 2, 16);
    p += __shfl_xor(p, 1, 16);
    if (s == 0) {
      float z   = xz[(size_t)t * (2 * D_INNER) + D_INNER + d];
      float sig = 1.f / (1.f + __expf(-z));
      y[(size_t)t * D_INNER + d] = (p + Dv * x) * (z * sig);
    }
  }
}

// ---------------------------------------------------------------------------
// Host-side orchestration
// ---------------------------------------------------------------------------
extern "C" void kernel_launch(void* const* d_in, const int* in_sizes, int n_in,
                              void* d_out, int out_size, void* d_ws, size_t ws_size,
                              hipStream_t stream) {
  const int*   ids    = (const int*)d_in[0];
  const float* emb    = (const float*)d_in[1];
  const float* in_w   = (const float*)d_in[2];
  const float* cw     = (const float*)d_in[3];
  const float* cb     = (const float*)d_in[4];
  const float* xp_w   = (const float*)d_in[5];
  const float* dt_w   = (const float*)d_in[6];
  const float* dt_b   = (const float*)d_in[7];
  const float* A_log  = (const float*)d_in[8];
  const float* Dp     = (const float*)d_in[9];
  const float* out_w  = (const float*)d_in[10];
  const float* norm_w = (const float*)d_in[11];
  const float* norm_b = (const float*)d_in[12];
  const float* fw     = (const float*)d_in[13];
  const float* fb     = (const float*)d_in[14];

  float* ws = (float*)d_ws;
  size_t off = 0;
  auto carve = [&](size_t n) {
    float* p = ws + off;
    off += (n + 63) & ~(size_t)63;   // 256B-aligned slices
    return p;
  };
  float* resid = carve((size_t)SEQ_L * D_MODEL);
  float* h     = carve((size_t)SEQ_L * D_MODEL);
  float* xz    = carve((size_t)SEQ_L * 2 * D_INNER);
  float* xc    = carve((size_t)SEQ_L * D_INNER);
  float* dblb  = carve((size_t)SEQ_L * DBL_N);
  float* dtr   = carve((size_t)SEQ_L * D_INNER);
  float* y     = carve((size_t)SEQ_L * D_INNER);
  float* hid   = carve((size_t)SEQ_L * D_MODEL);

  embed_kernel<<<SEQ_L, 256, 0, stream>>>(ids, emb, resid);

  for (int i = 0; i < NLAYERS; ++i) {
    add_ln_kernel<<<SEQ_L, 256, 0, stream>>>(resid, (i > 0) ? hid : nullptr,
                                             norm_w + (size_t)i * D_MODEL,
                                             norm_b + (size_t)i * D_MODEL, h);
    // xz = h @ in_w^T : M=512, N=7168, K=1792
    gemm_wmma<<<dim3(2 * D_INNER / BN, SEQ_L / BM), 256, 0, stream>>>(
        h, D_MODEL, in_w + (size_t)i * 2 * D_INNER * D_MODEL, D_MODEL,
        xz, 2 * D_INNER, 2 * D_INNER, D_MODEL, 0);
    conv_silu_kernel<<<dim3(D_INNER / 256, SEQ_L), 256, 0, stream>>>(
        xz, cw + (size_t)i * D_INNER * 4, cb + (size_t)i * D_INNER, xc);
    // dbl = xc @ xp_w^T : M=512, N=144, K=3584 (N needs bounds)
    gemm_wmma<<<dim3((DBL_N + BN - 1) / BN, SEQ_L / BM), 256, 0, stream>>>(
        xc, D_INNER, xp_w + (size_t)i * DBL_N * D_INNER, D_INNER,
        dblb, DBL_N, DBL_N, D_INNER, 1);
    // dt_raw = dbl[:, :112] @ dt_w^T : M=512, N=3584, K=112
    gemm_wmma<<<dim3(D_INNER / BN, SEQ_L / BM), 256, 0, stream>>>(
        dblb, DBL_N, dt_w + (size_t)i * D_INNER * DT_RANK, DT_RANK,
        dtr, D_INNER, D_INNER, DT_RANK, 0);
    scan_kernel<<<D_INNER / 16, 256, 0, stream>>>(
        dtr, dt_b + (size_t)i * D_INNER, dblb, xc, xz,
        A_log + (size_t)i * D_INNER * D_STATE, Dp + (size_t)i * D_INNER, y);
    // hidden = y @ out_w^T : M=512, N=1792, K=3584
    gemm_wmma<<<dim3(D_MODEL / BN, SEQ_L / BM), 256, 0, stream>>>(
        y, D_INNER, out_w + (size_t)i * D_MODEL * D_INNER, D_INNER,
        hid, D_MODEL, D_MODEL, D_INNER, 0);
  }

  add_ln_kernel<<<SEQ_L, 256, 0, stream>>>(resid, hid, fw, fb, (float*)d_out);
}